// VSA_73375221285089
// MI455X (gfx1250) — compile-verified
//
#include <hip/hip_runtime.h>

typedef __attribute__((ext_vector_type(2))) float v2f;
typedef __attribute__((ext_vector_type(8))) float v8f;

#define HW    16384   // h*w = 128*128
#define IMG_W 128
#define BCN   32      // b*c
#define CIN   64      // ah*aw input channels
#define CMID  96      // half*3
#define AN2   32      // half
#define TLS_ROWSTRIDE 512        // 4 rows * 128 cols per channel in LDS
#define TLS_FLOATS (48 * TLS_ROWSTRIDE)  // one 48-channel half-tile
#define KCHUNKS 16               // split-K factor for the attention GEMM

static __device__ __forceinline__ v8f wmma4(v2f a, v2f b, v8f c) {
  // V_WMMA_F32_16X16X4_F32 : D = A(16x4,f32) * B(4x16,f32) + C(16x16,f32)
  return __builtin_amdgcn_wmma_f32_16x16x4_f32(
      /*neg_a=*/false, a, /*neg_b=*/false, b,
      /*c_mod=*/(short)0, c, /*reuse_a=*/false, /*reuse_b=*/false);
}

// ---------------- K0: zero the norm accumulators + split-K S buffer ----------
__global__ void k_zero(float* __restrict__ p, int n) {
  int i = blockIdx.x * 256 + threadIdx.x;
  if (i < n) p[i] = 0.f;
}

// ---------------- K1: fused pointwise(WMMA) + depthwise 3x3 + q/k norms ------
// Block = 2 output rows (256 px) of one image, all 96 mid channels in 2 halves.
// t tile (4 rows incl. halo, 48 ch) lives only in LDS — never touches HBM.
// grid (64, BCN), block 256 (8 waves), dynamic LDS = 48*512*4 = 96 KB
__global__ void k_fused(const float* __restrict__ x,
                        const float* __restrict__ pw,
                        const float* __restrict__ dw,
                        float* __restrict__ qkv,
                        float* __restrict__ gsums) {
  extern __shared__ float tls[];          // [48][4][128]
  __shared__ float dwls[CMID * 9];
  __shared__ float chsums[2 * AN2];

  const int tid  = threadIdx.x;
  const int lane = tid & 31, wave = tid >> 5;
  const int lo = lane & 15, hi = lane >> 4;
  const int img = blockIdx.y;
  const int r0  = blockIdx.x * 2;         // first of the 2 output rows
  const float* __restrict__ X = x + (size_t)img * CIN * HW;
  float* __restrict__ QKV = qkv + (size_t)img * CMID * HW;

  for (int i = tid; i < CMID * 9; i += 256) dwls[i] = dw[i];
  if (tid < 2 * AN2) chsums[tid] = 0.f;
  __syncthreads();

  const int orow = tid >> 7;              // this thread's output row (0/1)
  const int col  = tid & (IMG_W - 1);     // this thread's column

  for (int half_ = 0; half_ < 2; ++half_) {
    const int cbase = half_ * 48;

    // ---- pointwise: 32 n-tiles (4 rows x 8 col-tiles) x 3 m-tiles, WMMA ----
    for (int nt = wave; nt < 32; nt += 8) {
      const int lrow = nt >> 3;                 // 0..3 (row within LDS tile)
      const int colbase = (nt & 7) * 16;
      const int rr = r0 - 1 + lrow;             // image row (may be out of range)
      if (rr < 0 || rr >= IMG_W) {
        // zero halo rows so depthwise needs no row-bound branches
        for (int i = lane; i < 48 * 16; i += 32)
          tls[(i >> 4) * TLS_ROWSTRIDE + lrow * IMG_W + colbase + (i & 15)] = 0.f;
      } else {
        const int pix = rr * IMG_W + colbase;
        v8f acc[3] = {};
        for (int k0 = 0; k0 < CIN; k0 += 4) {
          v2f b;                                   // B 4x16: (K=k0+2hi(+1), N=pix+lo)
          b.x = X[(size_t)(k0 + 2 * hi) * HW + pix + lo];
          b.y = X[(size_t)(k0 + 2 * hi + 1) * HW + pix + lo];
#pragma unroll
          for (int m = 0; m < 3; ++m) {
            v2f a;                                 // A 16x4 of pw_w (hot in WGP$)
            const float* arow = pw + (cbase + m * 16 + lo) * CIN + k0 + 2 * hi;
            a.x = arow[0];
            a.y = arow[1];
            acc[m] = wmma4(a, b, acc[m]);
          }
        }
#pragma unroll
        for (int m = 0; m < 3; ++m)
#pragma unroll
          for (int r = 0; r < 8; ++r)
            tls[(m * 16 + r + 8 * hi) * TLS_ROWSTRIDE + lrow * IMG_W + colbase + lo] =
                acc[m][r];
      }
    }
    __syncthreads();

    // ---- depthwise 3x3 from LDS; fused sum-of-squares for q,k channels ----
    for (int lc = 0; lc < 48; ++lc) {
      const int ch = cbase + lc;
      const float* __restrict__ tch = tls + lc * TLS_ROWSTRIDE + (1 + orow) * IMG_W + col;
      const float* __restrict__ wch = dwls + ch * 9;
      float s = 0.f;
#pragma unroll
      for (int dy = -1; dy <= 1; ++dy) {
        const float* rowp = tch + dy * IMG_W;
        if (col > 0)         s += wch[(dy + 1) * 3 + 0] * rowp[-1];
                             s += wch[(dy + 1) * 3 + 1] * rowp[0];
        if (col < IMG_W - 1) s += wch[(dy + 1) * 3 + 2] * rowp[1];
      }
      QKV[(size_t)ch * HW + (r0 + orow) * IMG_W + col] = s;
      if (ch < 2 * AN2) {
        float v = s * s;                       // wave-level tree reduction
#pragma unroll
        for (int off = 16; off > 0; off >>= 1) v += __shfl_xor(v, off, 32);
        if (lane == 0) atomicAdd(&chsums[ch], v);
      }
    }
    __syncthreads();                           // LDS tile reused by next half
  }

  if (tid < 2 * AN2) atomicAdd(&gsums[img * 64 + tid], chsums[tid]);
}

// ---------------- K2a: split-K partial S = q k^T -----------------------------
// grid (KCHUNKS, BCN), block 128 (4 waves = 2x2 tiles of 16x16).
// Each block reduces a 1024-pixel K-chunk with 4 independent accumulator
// chains per wave (breaks the serial WMMA RAW chain), then atomically
// combines its 32x32 partial into Sbuf.
__global__ void k_att_partial(const float* __restrict__ qkv,
                              float* __restrict__ Sbuf) {
  const int lane = threadIdx.x & 31, wave = threadIdx.x >> 5;
  const int lo = lane & 15, hi = lane >> 4;
  const int img = blockIdx.y;
  const float* __restrict__ Q = qkv + (size_t)img * CMID * HW;
  const float* __restrict__ Km = Q + (size_t)AN2 * HW;
  const int m0 = (wave >> 1) * 16, n0 = (wave & 1) * 16;
  const int kbase = blockIdx.x * (HW / KCHUNKS);

  const float* __restrict__ qrow = Q + (size_t)(m0 + lo) * HW + 2 * hi;
  const float* __restrict__ krow = Km + (size_t)(n0 + lo) * HW + 2 * hi;

  v8f acc0 = {}, acc1 = {}, acc2 = {}, acc3 = {};
  for (int kk = kbase; kk < kbase + HW / KCHUNKS; kk += 16) {
    v2f a, b;
    a.x = qrow[kk];      a.y = qrow[kk + 1];
    b.x = krow[kk];      b.y = krow[kk + 1];
    acc0 = wmma4(a, b, acc0);
    a.x = qrow[kk + 4];  a.y = qrow[kk + 5];
    b.x = krow[kk + 4];  b.y = krow[kk + 5];
    acc1 = wmma4(a, b, acc1);
    a.x = qrow[kk + 8];  a.y = qrow[kk + 9];
    b.x = krow[kk + 8];  b.y = krow[kk + 9];
    acc2 = wmma4(a, b, acc2);
    a.x = qrow[kk + 12]; a.y = qrow[kk + 13];
    b.x = krow[kk + 12]; b.y = krow[kk + 13];
    acc3 = wmma4(a, b, acc3);
  }
  v8f acc = (acc0 + acc1) + (acc2 + acc3);

  float* __restrict__ Sp = Sbuf + (size_t)img * (AN2 * AN2);
#pragma unroll
  for (int r = 0; r < 8; ++r)
    atomicAdd(&Sp[(m0 + r + 8 * hi) * AN2 + n0 + lo], acc[r]);
}

// ---------------- K2b: scale + softmax + fold proj_w -> W --------------------
// grid (BCN), block 128
__global__ void k_att_final(const float* __restrict__ Sbuf,
                            const float* __restrict__ sums,
                            const float* __restrict__ proj,
                            float* __restrict__ Wbuf) {
  const int img = blockIdx.x;
  __shared__ float S[AN2][AN2 + 1];
  __shared__ float rq[AN2], rk[AN2];

  if (threadIdx.x < 32) {
    rq[threadIdx.x] = 1.f / fmaxf(sqrtf(sums[img * 64 + threadIdx.x]), 1e-12f);
  } else if (threadIdx.x < 64) {
    int d = threadIdx.x - 32;
    rk[d] = 1.f / fmaxf(sqrtf(sums[img * 64 + 32 + d]), 1e-12f);
  }
  __syncthreads();
  const float* __restrict__ Sp = Sbuf + (size_t)img * (AN2 * AN2);
  for (int idx = threadIdx.x; idx < AN2 * AN2; idx += 128) {
    int row = idx >> 5, col = idx & 31;
    S[row][col] = Sp[idx] * rq[row] * rk[col];
  }
  __syncthreads();
  if (threadIdx.x < 32) {  // softmax over each 32-wide row
    int row = threadIdx.x;
    float mx = -3.4e38f;
    for (int j = 0; j < 32; ++j) mx = fmaxf(mx, S[row][j]);
    float sum = 0.f;
    for (int j = 0; j < 32; ++j) {
      float e = __expf(S[row][j] - mx);
      S[row][j] = e;
      sum += e;
    }
    float inv = 1.f / sum;
    for (int j = 0; j < 32; ++j) S[row][j] *= inv;
  }
  __syncthreads();
  // Fold projection: W[o][d] = sum_c proj[o][c] * att[c][d]   (64x32, tiny)
  for (int idx = threadIdx.x; idx < 64 * 32; idx += 128) {
    int o = idx >> 5, d = idx & 31;
    float s = 0.f;
    for (int cc = 0; cc < 32; ++cc) s += proj[o * 32 + cc] * S[cc][d];
    Wbuf[(size_t)img * 2048 + idx] = s;
  }
}

// ---------------- K3: out = W @ v  (M=64, K=32, N=HW) via WMMA ----------------
// grid (HW/64, BCN), block 128 (4 waves, each 16-pixel N tile, all 4 M tiles)
__global__ void k_out(const float* __restrict__ qkv,
                      const float* __restrict__ Wbuf,
                      float* __restrict__ out) {
  const int lane = threadIdx.x & 31, wave = threadIdx.x >> 5;
  const int lo = lane & 15, hi = lane >> 4;
  const int img = blockIdx.y;
  const int n0 = blockIdx.x * 64 + wave * 16;
  const float* __restrict__ V = qkv + (size_t)img * CMID * HW + (size_t)(2 * AN2) * HW;
  const float* __restrict__ W = Wbuf + (size_t)img * 2048;
  float* __restrict__ O = out + (size_t)img * CIN * HW;

  v8f acc[4] = {};
  for (int k0 = 0; k0 < 32; k0 += 4) {
    v2f b;
    b.x = V[(size_t)(k0 + 2 * hi) * HW + n0 + lo];
    b.y = V[(size_t)(k0 + 2 * hi + 1) * HW + n0 + lo];
#pragma unroll
    for (int m = 0; m < 4; ++m) {
      v2f a;
      const float* wr = W + (m * 16 + lo) * 32 + k0 + 2 * hi;
      a.x = wr[0];
      a.y = wr[1];
      acc[m] = wmma4(a, b, acc[m]);
    }
  }
#pragma unroll
  for (int m = 0; m < 4; ++m)
#pragma unroll
    for (int r = 0; r < 8; ++r)
      // write-once output: non-temporal so it doesn't evict qkv from L2
      __builtin_nontemporal_store(acc[m][r],
          &O[(size_t)(m * 16 + r + 8 * hi) * HW + n0 + lo]);
}

extern "C" void kernel_launch(void* const* d_in, const int* in_sizes, int n_in,
                              void* d_out, int out_size, void* d_ws, size_t ws_size,
                              hipStream_t stream) {
  const float* x    = (const float*)d_in[0];  // (2,16,8,8,128,128) -> (32,64,HW)
  const float* pw   = (const float*)d_in[1];  // (96,64)
  const float* dw   = (const float*)d_in[2];  // (96,1,3,3)
  const float* proj = (const float*)d_in[3];  // (64,32)
  float* out = (float*)d_out;                 // (32,64,HW)

  // Workspace layout (~202 MB total)
  float* qkv  = (float*)d_ws;                      // BCN*CMID*HW
  float* sums = qkv + (size_t)BCN * CMID * HW;     // BCN*64 (q,k sum-of-squares)
  float* Sbuf = sums + (size_t)BCN * 64;           // BCN*32*32 split-K partials
  float* Wbuf = Sbuf + (size_t)BCN * AN2 * AN2;    // BCN*64*32
  const int nzero = BCN * 64 + BCN * AN2 * AN2;    // sums + Sbuf (contiguous)

  k_zero<<<dim3((nzero + 255) / 256), 256, 0, stream>>>(sums, nzero);
  k_fused<<<dim3(IMG_W / 2, BCN), 256, TLS_FLOATS * sizeof(float), stream>>>(
      x, pw, dw, qkv, sums);
  k_att_partial<<<dim3(KCHUNKS, BCN), 128, 0, stream>>>(qkv, Sbuf);
  k_att_final<<<dim3(BCN), 128, 0, stream>>>(Sbuf, sums, proj, Wbuf);
  k_out<<<dim3(HW / 64, BCN), 128, 0, stream>>>(qkv, Wbuf, out);
}